// ChamferLoss_27204322853516
// MI455X (gfx1250) — compile-verified
//
#include <hip/hip_runtime.h>
#include <hip/hip_bf16.h>

typedef __attribute__((ext_vector_type(2))) float v2f;
typedef __attribute__((ext_vector_type(8))) float v8f;

#define B_  16
#define N_  4096
#define M_  4096
#define NM_ (B_ * N_)   // 65536, also == B_*M_
#define RT_ 4           // row-tiles per wave (64 rows)
#define ROWS_PER_BLOCK_ (8 * RT_ * 16)   // 8 waves * 64 rows = 512
#define NBLK_ ((N_ / ROWS_PER_BLOCK_) * B_)   // 8 * 16 = 128

// ---------------------------------------------------------------------------
// Kernel 1: pack gts -> (x,y,z,xx), preds -> (-2x,-2y,-2z,yy), init colmin
// ---------------------------------------------------------------------------
__global__ __launch_bounds__(256)
void chamfer_pack(const float* __restrict__ gts, const float* __restrict__ preds,
                  float4* __restrict__ gX4, float4* __restrict__ gY4,
                  unsigned* __restrict__ gcolmin) {
    int i = blockIdx.x * blockDim.x + threadIdx.x;   // 0 .. 65535
    float x = gts[i * 3 + 0], y = gts[i * 3 + 1], z = gts[i * 3 + 2];
    gX4[i] = make_float4(x, y, z, x * x + y * y + z * z);
    float px = preds[i * 3 + 0], py = preds[i * 3 + 1], pz = preds[i * 3 + 2];
    gY4[i] = make_float4(-2.f * px, -2.f * py, -2.f * pz, px * px + py * py + pz * pz);
    gcolmin[i] = 0xFFFFFFFFu;                        // +inf under monotone key encoding
}

// Order-preserving float -> uint key (unsigned compare == float compare)
__device__ __forceinline__ unsigned f2key(float f) {
    unsigned b = __float_as_uint(f);
    return (b & 0x80000000u) ? ~b : (b | 0x80000000u);
}
__device__ __forceinline__ float key2f(unsigned k) {
    return __uint_as_float((k & 0x80000000u) ? (k & 0x7fffffffu) : ~k);
}

// ---------------------------------------------------------------------------
// Kernel 2: main. Block = 256 thr (8 waves), grid = (N/512, B).
// Wave owns 64 rows (4 WMMA row-tiles); inner loop covers 2 col-tiles:
// 8 x v_wmma_f32_16x16x4_f32 per iteration (2048 pairs).
// D = A(16x4) * B(4x16) + C:  A=[x0,x1,x2,1], B=[-2y0,-2y1,-2y2,yy], C=xx
// ---------------------------------------------------------------------------
__global__ __launch_bounds__(256)
void chamfer_main(const float4* __restrict__ gX4, const float4* __restrict__ gY4,
                  unsigned* __restrict__ gcolmin, float* __restrict__ partials) {
    const int  b    = blockIdx.y;
    const int  lane = threadIdx.x & 31;
    const int  w    = threadIdx.x >> 5;
    const int  l15  = lane & 15;
    const bool hi   = lane >= 16;          // lanes 16..31 hold K=2,3 of A/B frags
    const int  rbase = blockIdx.x * ROWS_PER_BLOCK_ + w * (RT_ * 16);

    const float4* Xb = gX4 + b * N_;
    const float4* Yb = gY4 + b * M_;

    const float INF = __builtin_inff();
    const int cbase = hi ? 8 : 0;

    // Per-row-tile fragments (loop invariant)
    v2f a[RT_];
    v8f c[RT_];
    v8f rmin[RT_];
#pragma unroll
    for (int t = 0; t < RT_; ++t) {
        float4 xv = Xb[rbase + t * 16 + l15];
        a[t].x = hi ? xv.z : xv.x;
        a[t].y = hi ? 1.0f : xv.y;
#pragma unroll
        for (int k = 0; k < 8; ++k) {
            c[t][k]    = __shfl(xv.w, cbase + k, 32);
            rmin[t][k] = INF;
        }
    }

    unsigned* colb = gcolmin + b * M_;

    for (int m0 = 0; m0 < M_; m0 += 32) {
        // Two B fragments: lane<16 -> (-2y0,-2y1); lane>=16 -> (-2y2, yy)
        float4 yA = Yb[m0 + l15];
        float4 yB = Yb[m0 + 16 + l15];
        v2f bA, bB;
        bA.x = hi ? yA.z : yA.x;  bA.y = hi ? yA.w : yA.y;
        bB.x = hi ? yB.z : yB.x;  bB.y = hi ? yB.w : yB.y;

        v8f dA[RT_], dB[RT_];
#pragma unroll
        for (int t = 0; t < RT_; ++t)
            dA[t] = __builtin_amdgcn_wmma_f32_16x16x4_f32(
                false, a[t], false, bA, (short)0, c[t], false, false);
#pragma unroll
        for (int t = 0; t < RT_; ++t)
            dB[t] = __builtin_amdgcn_wmma_f32_16x16x4_f32(
                false, a[t], false, bB, (short)0, c[t], false, false);

        // Row mins: one v_min3 per accumulator element (folds both col-tiles)
#pragma unroll
        for (int t = 0; t < RT_; ++t)
#pragma unroll
            for (int k = 0; k < 8; ++k)
                rmin[t][k] = fminf(rmin[t][k], fminf(dA[t][k], dB[t][k]));

        // Column mins over the 64 rows this wave owns (min3 chains)
        float cA = INF, cB = INF;
#pragma unroll
        for (int t = 0; t < RT_; ++t)
#pragma unroll
            for (int k = 0; k < 8; ++k) {
                cA = fminf(cA, dA[t][k]);
                cB = fminf(cB, dB[t][k]);
            }
        // combine the two 16-lane halves (rows k vs k+8 of each tile)
        cA = fminf(cA, __shfl_xor(cA, 16, 32));
        cB = fminf(cB, __shfl_xor(cB, 16, 32));
        if (!hi) {
            atomicMin(&colb[m0 + lane],      f2key(cA));  // global_atomic_min_u32
            atomicMin(&colb[m0 + 16 + lane], f2key(cB));
        }
    }

    // Row-min: reduce across the 16 columns held by each half-wave
#pragma unroll
    for (int mask = 1; mask <= 8; mask <<= 1)
#pragma unroll
        for (int t = 0; t < RT_; ++t)
#pragma unroll
            for (int k = 0; k < 8; ++k)
                rmin[t][k] = fminf(rmin[t][k], __shfl_xor(rmin[t][k], mask, 32));

    // Sum of this wave's 64 row-mins (half0 rows k / half1 rows k+8)
    float s = 0.f;
#pragma unroll
    for (int t = 0; t < RT_; ++t)
#pragma unroll
        for (int k = 0; k < 8; ++k) s += rmin[t][k];
    s += __shfl_xor(s, 16, 32);            // add the other half's rows

    __shared__ float wsum[8];
    if (lane == 0) wsum[w] = s;
    __syncthreads();
    if (threadIdx.x == 0) {
        float t = 0.f;
#pragma unroll
        for (int i = 0; i < 8; ++i) t += wsum[i];
        partials[blockIdx.y * gridDim.x + blockIdx.x] = t;   // 128 partials
    }
}

// ---------------------------------------------------------------------------
// Kernel 3: deterministic final reduction -> scalar loss
// ---------------------------------------------------------------------------
__global__ __launch_bounds__(256)
void chamfer_final(const unsigned* __restrict__ gcolmin,
                   const float* __restrict__ partials, int nblk,
                   float* __restrict__ out) {
    __shared__ float red[256];
    int t = threadIdx.x;
    float s = 0.f;
    for (int i = t; i < NM_; i += 256) s += key2f(gcolmin[i]);   // loss_1 numerator
    for (int i = t; i < nblk; i += 256) s += partials[i];        // loss_2 numerator
    red[t] = s;
    __syncthreads();
    for (int off = 128; off > 0; off >>= 1) {
        if (t < off) red[t] += red[t + off];
        __syncthreads();
    }
    if (t == 0) out[0] = red[0] * (1.0f / (float)NM_);  // B*M == B*N == 65536
}

// ---------------------------------------------------------------------------
extern "C" void kernel_launch(void* const* d_in, const int* in_sizes, int n_in,
                              void* d_out, int out_size, void* d_ws, size_t ws_size,
                              hipStream_t stream) {
    const float* gts   = (const float*)d_in[0];   // [16,4096,3]
    const float* preds = (const float*)d_in[1];   // [16,4096,3]
    float* out = (float*)d_out;

    // workspace layout
    char* ws = (char*)d_ws;
    float4*   gX4      = (float4*)ws;                       // 65536 * 16B = 1 MB
    float4*   gY4      = gX4 + NM_;                         // 1 MB
    unsigned* gcolmin  = (unsigned*)(gY4 + NM_);            // 256 KB
    float*    partials = (float*)(gcolmin + NM_);           // 512 B
    (void)in_sizes; (void)n_in; (void)out_size; (void)ws_size;

    chamfer_pack <<<NM_ / 256, 256, 0, stream>>>(gts, preds, gX4, gY4, gcolmin);
    chamfer_main <<<dim3(N_ / ROWS_PER_BLOCK_, B_), 256, 0, stream>>>(gX4, gY4, gcolmin, partials);
    chamfer_final<<<1, 256, 0, stream>>>(gcolmin, partials, NBLK_, out);
}